// NodeAttention_66348654788873
// MI455X (gfx1250) — compile-verified
//
#include <hip/hip_runtime.h>
#include <cstdint>
#include <cstddef>

typedef __attribute__((ext_vector_type(2))) float v2f;
typedef __attribute__((ext_vector_type(8))) float v8f;

#define D_DIM 256

// ---------------------------------------------------------------------------
// Kernel 1: diag[n] = sigmoid(dot(x[n,:], w) + b), 16 rows per wave via
// V_WMMA_F32_16X16X4_F32.  A-matrix = 16 rows of x (lanes 0-15 carry M,
// lane-half selects K pair {0,1}/{2,3}); B-matrix = w[k] broadcast across all
// 16 columns (B K-striping mirrors A).  Every column of D then holds the same
// GEMV result; C/D layout: VGPR r = row r (lanes 0-15) / row 8+r (lanes 16-31).
// ---------------------------------------------------------------------------
__global__ __launch_bounds__(256) void gemv_sigmoid_wmma(
    const float* __restrict__ x, const float* __restrict__ w,
    const float* __restrict__ b, float* __restrict__ diag,
    int n, int ntiles)
{
    __shared__ __align__(16) float sw[D_DIM];
    for (int i = threadIdx.x; i < D_DIM; i += blockDim.x) sw[i] = w[i];
    __syncthreads();

    const int wave = threadIdx.x >> 5;
    const int lane = threadIdx.x & 31;
    const int tile = blockIdx.x * (blockDim.x >> 5) + wave;
    if (tile >= ntiles) return;                 // wave-uniform: EXEC stays all-1s

    const int m     = lane & 15;
    const int khalf = (lane >> 4) << 1;         // 0 for lanes 0-15, 2 for 16-31
    int mrow = tile * 16 + m;
    if (mrow >= n) mrow = n - 1;                // clamp: duplicate work, guarded write
    const float* xrow = x + (size_t)mrow * D_DIM + khalf;

    v8f acc = {};
    #pragma unroll 4
    for (int kb = 0; kb < D_DIM; kb += 4) {
        v2f a  = *(const v2f*)(xrow + kb);          // x[mrow][kb+khalf .. +1]
        v2f bv = *(const v2f*)(&sw[kb + khalf]);    // w[kb+khalf .. +1] (bcast cols)
        acc = __builtin_amdgcn_wmma_f32_16x16x4_f32(
            false, a, false, bv, (short)0, acc, false, false);
    }

    const float bias = b[0];
    if ((lane & 15) == 0) {                     // lanes 0 and 16
        const int base = tile * 16 + (lane >> 4) * 8;
        #pragma unroll
        for (int r = 0; r < 8; ++r) {
            const int row = base + r;
            if (row < n) {
                const float z = acc[r] + bias;
                // sigmoid via v_rcp_f32 (~1 ulp) instead of full IEEE div chain
                diag[row] = __builtin_amdgcn_rcpf(1.0f + __expf(-z));
            }
        }
    }
}

// ---------------------------------------------------------------------------
// Kernel 2: zero the degree array (ws is not re-poisoned between replays).
// ---------------------------------------------------------------------------
__global__ void zero_f32(float* __restrict__ p, int n)
{
    const int i = blockIdx.x * blockDim.x + threadIdx.x;
    if (i < n) p[i] = 0.0f;
}

// ---------------------------------------------------------------------------
// Kernel 3: deg[col[e]] += 1.0f.  All addends are 1.0f and per-node counts
// are < 2^24, so the float sum is exact and order-independent (deterministic).
// deg is 400 KB -> atomics land in L2 (192 MB), HBM traffic is just col[].
// ---------------------------------------------------------------------------
__global__ void degree_scatter(const int* __restrict__ col,
                               float* __restrict__ deg, int e)
{
    const int i = blockIdx.x * blockDim.x + threadIdx.x;
    if (i < e) atomicAdd(&deg[col[i]], 1.0f);
}

// ---------------------------------------------------------------------------
// Kernel 4: deg_inv = 1.0f / deg, per node (matches reference: deg.pow(-1)
// computed once per node; deg==0 -> +inf, IEEE divide).  100k divides total
// instead of 3.2M in the edge pass.
// ---------------------------------------------------------------------------
__global__ void reciprocal_f32(float* __restrict__ p, int n)
{
    const int i = blockIdx.x * blockDim.x + threadIdx.x;
    if (i < n) p[i] = 1.0f / p[i];
}

// ---------------------------------------------------------------------------
// Kernel 5: out = [edge_index as float (2E) | edge_attr*diag[col]*deg_inv[row]].
// diag/deg_inv gathers are L2-resident; everything else streams coalesced.
// Pure loads + 2 muls + 3 stores per edge.
// ---------------------------------------------------------------------------
__global__ void finalize_edges(const int* __restrict__ ei,
                               const float* __restrict__ ea,
                               const float* __restrict__ diag,
                               const float* __restrict__ deg_inv,
                               float* __restrict__ out, int e)
{
    const int i = blockIdx.x * blockDim.x + threadIdx.x;
    if (i < e) {
        const int r = ei[i];         // row endpoint
        const int c = ei[e + i];     // col endpoint
        out[i]         = (float)r;
        out[e + i]     = (float)c;
        out[2 * e + i] = ea[i] * diag[c] * deg_inv[r];
    }
}

extern "C" void kernel_launch(void* const* d_in, const int* in_sizes, int n_in,
                              void* d_out, int out_size, void* d_ws, size_t ws_size,
                              hipStream_t stream)
{
    const float* x  = (const float*)d_in[0];   // [N, 256]
    const int*   ei = (const int*)d_in[1];     // [2, E] flattened
    const float* ea = (const float*)d_in[2];   // [E]
    const float* w  = (const float*)d_in[3];   // [256]
    const float* b  = (const float*)d_in[4];   // [1]
    float* out = (float*)d_out;                // [2E + E] floats

    const int d = in_sizes[3];                 // 256
    const int n = in_sizes[0] / d;             // 100000
    const int e = in_sizes[2];                 // 3200000

    float* diag = (float*)d_ws;                // [n] floats
    float* deg  = diag + n;                    // [n] floats (becomes deg_inv)

    // deg = 0
    zero_f32<<<(n + 255) / 256, 256, 0, stream>>>(deg, n);
    // deg[col] += 1  (col = second row of edge_index)
    degree_scatter<<<(e + 255) / 256, 256, 0, stream>>>(ei + e, deg, e);
    // deg -> 1/deg (per node, IEEE divide; 0 -> inf as in reference)
    reciprocal_f32<<<(n + 255) / 256, 256, 0, stream>>>(deg, n);
    // diag = sigmoid(x @ w + b)   (WMMA f32 16x16x4 GEMV, 16 rows per wave)
    const int ntiles = (n + 15) / 16;
    const int wavesPerBlock = 8;               // 256 threads = 8 wave32
    const int blocks = (ntiles + wavesPerBlock - 1) / wavesPerBlock;
    gemv_sigmoid_wmma<<<blocks, 32 * wavesPerBlock, 0, stream>>>(x, w, b, diag, n, ntiles);
    // out = [edge_index (as float), deg_inv[row] * edge_attr * diag[col]]
    finalize_edges<<<(e + 255) / 256, 256, 0, stream>>>(ei, ea, diag, deg, out, e);
}